// Model_54795192763123
// MI455X (gfx1250) — compile-verified
//
#include <hip/hip_runtime.h>

#define ROIS 2048
#define ROWF 64            // floats per tube row (SD*4)
#define THRESH 0.25f
#define CHUNK 64           // rows per block (gridDim.y = ROIS/CHUNK)

typedef __attribute__((ext_vector_type(2))) float v2f;
typedef __attribute__((ext_vector_type(8))) float v8f;

// min/max for known-non-negative, NaN-free floats: the float ordering equals
// the signed-int ordering of the bit patterns, so these lower to single
// v_max_i32/v_min_i32 with no canonicalize (v_max_num x,x) overhead.
__device__ __forceinline__ float fmax_nn(float a, float b) {
    return __int_as_float(max(__float_as_int(a), __float_as_int(b)));
}
__device__ __forceinline__ float fmin_nn(float a, float b) {
    return __int_as_float(min(__float_as_int(a), __float_as_int(b)));
}
// fmaxf(w, 0.0f) for possibly-negative w: negative floats (incl. -0.0) have
// MSB set -> negative as signed int -> clamped to 0. Bit-exact vs fmaxf.
__device__ __forceinline__ float relu_nn(float w) {
    return __int_as_float(max(__float_as_int(w), 0));
}

// ---------------------------------------------------------------------------
// Zero the per-step argmax key arrays (31 x 2048 u64) in d_ws.
// ---------------------------------------------------------------------------
__global__ void init_keys_kernel(unsigned long long* __restrict__ keys, int n) {
    int i = blockIdx.x * blockDim.x + threadIdx.x;
    if (i < n) keys[i] = 0ull;
}

// ---------------------------------------------------------------------------
// Forward step s (s = 0..30):
//   keys_cur[j] = max over rows i of pack(score_prev[i] + iou(i,j), i)
//                 restricted to iou > 0.25, else stays 0 ("no valid link").
// score_prev is reconstructed on the fly from keys_prev + actioness[s]
// (pure function -> every block recomputes it, no cross-block sync needed).
//
// Block = 256 threads = 8 waves; covers 128 columns (blockIdx.x) x 64 rows
// (blockIdx.y). Each wave owns 16 columns and walks 4 row-tiles of 16 rows.
// Per tile, two V_WMMA_F32_16X16X4_F32 ops produce, in the 16x16 C/D layout
// (lane<16 -> rows m=0..7 in c[0..7], lane>=16 -> m=8..15, col = lane&15):
//   c1[m,n] = area_a[m] + area_b[n]     (A1=[areaA,1], B=[1;areaB])
//   c2[m,n] = score_prev[m]             (A2=[sPrev,0], same B operand)
// Both are exact under RNE (x*1 and +0 are exact), so results match scalar.
// The IoU update loop is fully branchless (cndmask selects, no exec waves).
// ---------------------------------------------------------------------------
__global__ __launch_bounds__(256) void fwd_step_kernel(
    const float* __restrict__ p_tubes,
    const float* __restrict__ act,
    const unsigned long long* __restrict__ keys_prev,   // null for s == 0
    unsigned long long* __restrict__ keys_cur,
    int s)
{
    __shared__ float4 sBox[CHUNK];   // last_boxes[s][rbase + i]
    __shared__ float  sArea[CHUNK];  // area_a
    __shared__ float  sPrev[CHUNK];  // score_prev

    const int tid   = threadIdx.x;
    const int rbase = blockIdx.y * CHUNK;

    // ---- stage this row chunk into LDS --------------------------------
    if (tid < CHUNK) {
        const int row = rbase + tid;
        const float4 bb =
            *(const float4*)(p_tubes + ((size_t)s * ROIS + row) * ROWF + 60);
        sBox[tid]  = bb;
        sArea[tid] = (bb.z - bb.x + 1.0f) * (bb.w - bb.y + 1.0f);
        float sp = act[(size_t)s * ROIS + row];
        if (keys_prev) {
            unsigned long long k = keys_prev[row];
            if (k) sp += __uint_as_float((unsigned)(k >> 32));
        }
        sPrev[tid] = sp;
    }
    __syncthreads();

    const int  lane = tid & 31;
    const int  wave = tid >> 5;
    const bool lo   = lane < 16;
    const int  ln16 = lane & 15;

    // this lane's column (first_boxes[s+1][n])
    const int n = blockIdx.x * 128 + wave * 16 + ln16;
    const float4 cb =
        *(const float4*)(p_tubes + ((size_t)(s + 1) * ROIS + n) * ROWF);
    const float areaB = (cb.z - cb.x + 1.0f) * (cb.w - cb.y + 1.0f);

    // B matrix (4x16, KxN): B[0,n]=1, B[1,n]=areaB, B[2..3,*]=0
    // lanes 0-15 hold K=0 (VGPR0) / K=1 (VGPR1); lanes 16-31 hold K=2,3 -> 0.
    v2f bmat;
    bmat.x = lo ? 1.0f  : 0.0f;
    bmat.y = lo ? areaB : 0.0f;

    float best = -1.0f;   // any valid cand is > 0.25
    int   brow = 0;

#pragma unroll
    for (int tile = 0; tile < CHUNK / 16; ++tile) {
        const int tb = tile * 16;
        // Unconditional in-bounds LDS reads; select to zero on lanes 16-31.
        const float av = sArea[tb + ln16];
        const float sv = sPrev[tb + ln16];

        v2f amat1;                       // A1[m,*] = [areaA[m], 1, 0, 0]
        amat1.x = lo ? av   : 0.0f;
        amat1.y = lo ? 1.0f : 0.0f;
        v2f amat2;                       // A2[m,*] = [sPrev[m], 0, 0, 0]
        amat2.x = lo ? sv : 0.0f;
        amat2.y = 0.0f;

        v8f cz = {};
        // c1[m,n] = areaA[m] + areaB[n]
        v8f c1 = __builtin_amdgcn_wmma_f32_16x16x4_f32(
            false, amat1, false, bmat, (short)0, cz, false, false);
        // c2[m,n] = sPrev[m]  (same B operand)
        v8f c2 = __builtin_amdgcn_wmma_f32_16x16x4_f32(
            false, amat2, false, bmat, (short)0, cz, false, false);

        const int m0 = tb + (lo ? 0 : 8);
#pragma unroll
        for (int k = 0; k < 8; ++k) {
            const int ml = m0 + k;                 // local row index
            const float4 a4 = sBox[ml];
            float x1 = fmax_nn(a4.x, cb.x);        // coords are >= 0: int min/max
            float y1 = fmax_nn(a4.y, cb.y);
            float x2 = fmin_nn(a4.z, cb.z);
            float y2 = fmin_nn(a4.w, cb.w);
            float iw = relu_nn(x2 - x1 + 1.0f);
            float ih = relu_nn(y2 - y1 + 1.0f);
            float inter = iw * ih;
            float denom = c1[k] - inter;           // areaA + areaB - inter
            float iou   = inter / denom;           // IEEE div: argmax-exact vs ref
            float cand  = c2[k] + iou;             // sPrev[ml] + iou
            bool  upd   = (iou > THRESH) && (cand > best);
            best = upd ? cand : best;
            brow = upd ? (rbase + ml) : brow;
        }
    }

    // merge across blocks / across the two lanes sharing this column.
    // key encodes (score, first-index tie-break): scores are non-negative so
    // their IEEE bits are monotonic; 0x7FFFFFFF - row prefers smaller rows.
    if (best >= 0.0f) {
        unsigned long long key =
            ((unsigned long long)__float_as_uint(best) << 32) |
            (unsigned long long)(0x7FFFFFFFu - (unsigned)brow);
        atomicMax(keys_cur + n, key);
    }
}

// ---------------------------------------------------------------------------
// Decode keys: conn_scores + backward link tracing -> pos (2048 x 32 int32).
// ---------------------------------------------------------------------------
__global__ void backward_kernel(
    const unsigned long long* __restrict__ keys,
    const float* __restrict__ act,
    int* __restrict__ out_pos,        // 2048 x 32, row-major
    float* __restrict__ out_scores)   // 2048
{
    int j = blockIdx.x * blockDim.x + threadIdx.x;
    if (j >= ROIS) return;

    // final carry of the scan: score after step 30 for column j
    unsigned long long k30 = keys[30 * ROIS + j];
    float sc = act[(size_t)31 * ROIS + j];
    if (k30) sc += __uint_as_float((unsigned)(k30 >> 32));
    out_scores[j] = sc;

    int cur = j;
    out_pos[j * 32 + 31] = j;
    for (int s = 30; s >= 0; --s) {
        int prev = -1;
        if (cur >= 0) {
            unsigned long long k = keys[(size_t)s * ROIS + cur];
            prev = k ? (int)(0x7FFFFFFFu - (unsigned)(k & 0xFFFFFFFFu)) : -1;
        }
        out_pos[j * 32 + s] = prev;
        cur = prev;
    }
}

// ---------------------------------------------------------------------------
extern "C" void kernel_launch(void* const* d_in, const int* in_sizes, int n_in,
                              void* d_out, int out_size, void* d_ws, size_t ws_size,
                              hipStream_t stream) {
    (void)in_sizes; (void)n_in; (void)out_size; (void)ws_size;

    const float* p_tubes = (const float*)d_in[0];   // (32, 2048, 64) f32
    const float* act     = (const float*)d_in[1];   // (32, 2048) f32

    unsigned long long* keys = (unsigned long long*)d_ws;  // 31 x 2048 u64

    const int nkeys = 31 * ROIS;
    init_keys_kernel<<<(nkeys + 255) / 256, 256, 0, stream>>>(keys, nkeys);

    for (int s = 0; s < 31; ++s) {
        fwd_step_kernel<<<dim3(16, ROIS / CHUNK), 256, 0, stream>>>(
            p_tubes, act,
            s ? (keys + (size_t)(s - 1) * ROIS) : (const unsigned long long*)nullptr,
            keys + (size_t)s * ROIS, s);
    }

    int*   out_pos    = (int*)d_out;                       // 2048*32 int32
    float* out_scores = (float*)d_out + (size_t)ROIS * 32; // 2048 f32
    backward_kernel<<<ROIS / 256, 256, 0, stream>>>(keys, act, out_pos, out_scores);
}